// NestedOscillator_23270132810187
// MI455X (gfx1250) — compile-verified
//
#include <hip/hip_runtime.h>
#include <hip/hip_bf16.h>

// ---------------------------------------------------------------------------
// NestedOscillator on gfx1250 (MI455X).
//
// Kernel 1 (osc_scan): single-lane exact sequential scan (fp order matches the
//   jax.lax.scan reference: record-then-update, conditional fast-phase reset
//   on slow-phase wrap). Emits slow_t -> out[0..N), fast_t -> out[N..2N).
//   Fast path replaces jnp.mod with a conditional subtract that is provably
//   bit-identical for in-range operands (Sterbenz exactness), collapsing the
//   dependent chain to ~3 VALU ops/step. Generic fmod path kept as fallback.
// Kernel 2 (osc_scale_wmma): out[2N..3N) = slow * (1/2pi) computed on the
//   matrix pipe with chained V_WMMA_F32_16X16X4_F32. Each output element gets
//   exactly one non-zero product (s * slow) + exact zeros, so the result is
//   bit-identical to a scalar v_mul_f32 (RNE), but exercises the WMMA path.
// ---------------------------------------------------------------------------

typedef __attribute__((ext_vector_type(2))) float v2f;   // 16x4 f32 A/B operand: 2 VGPRs
typedef __attribute__((ext_vector_type(8))) float v8f;   // 16x16 f32 C/D operand: 8 VGPRs

#define TWO_PI_F     6.28318530717958647692f
#define PI_F         3.14159265358979323846f
#define INV_TWO_PI_F 0.15915494309189533577f
#define DT_F         0.001f

// Python-style positive mod (jnp.mod): result takes sign of divisor.
__device__ __forceinline__ float pmod(float x, float m) {
    float r = fmodf(x, m);            // fmod is exact; matches jnp.mod for x >= 0
    return (r < 0.0f) ? (r + m) : r;
}

// Bit-exact jnp.mod(x + d, 2pi) for x in [0, 2pi), d in [0, 2pi):
// t < 4pi strictly; for t in [2pi, 4pi) Sterbenz => t - 2pi exact == fmod.
__device__ __forceinline__ float wrap_add(float x, float d) {
    float t = x + d;
    return (t >= TWO_PI_F) ? (t - TWO_PI_F) : t;
}

struct alignas(16) F4 { float v[4]; };

__global__ void osc_scan_kernel(const float* __restrict__ p_log_slow,
                                const float* __restrict__ p_log_fast,
                                const float* __restrict__ p_reset,
                                float* __restrict__ out, int N) {
    if (threadIdx.x != 0 || blockIdx.x != 0) return;

    // Loop-invariant scalars (two RNE roundings, same order as XLA: exp, *dt)
    const float dslow = expf(p_log_slow[0]) * DT_F;
    const float dfast = expf(p_log_fast[0]) * DT_F;
    const float keep  = 1.0f - p_reset[0];

    float slow = 0.0f, fast = 0.0f, prev = 0.0f;
    float* __restrict__ oS = out;       // slow_phases
    float* __restrict__ oF = out + N;   // fast_phases

    // Fast path legal iff deltas in [0, 2pi) and keep in [0, 1]; the range
    // proofs above then make cond-subtract / bare multiply bit-identical to
    // jnp.mod. NaN params fail these checks -> generic path.
    const bool fast_ok =
        (dslow >= 0.0f) && (dslow < TWO_PI_F) &&
        (dfast >= 0.0f) && (dfast < TWO_PI_F) &&
        (keep  >= 0.0f) && (keep  <= 1.0f);

    int t = 0;
    if (fast_ok && ((N & 3) == 0)) {
        // ~3 dependent VALU ops per step; 4-step buffering -> 16B stores
        // (base is 256B aligned, N % 4 == 0 keeps out+N 16B aligned).
        for (; t < N; t += 4) {
            F4 s4, f4;
#pragma unroll
            for (int u = 0; u < 4; ++u) {
                s4.v[u] = slow;                     // record BEFORE update
                f4.v[u] = fast;
                bool crossed = (prev > PI_F) && (slow < PI_F);  // trough crossing
                float ns = wrap_add(slow, dslow);
                float nf = wrap_add(fast, dfast);
                nf = crossed ? nf * keep : nf;      // product in [0,2pi): mod = id
                prev = slow; slow = ns; fast = nf;
            }
            *(F4*)(oS + t) = s4;
            *(F4*)(oF + t) = f4;
        }
    }
    for (; t < N; ++t) {                // generic / tail path (full fmod)
        oS[t] = slow;
        oF[t] = fast;
        bool crossed = (prev > PI_F) && (slow < PI_F);
        float ns = pmod(slow + dslow, TWO_PI_F);
        float nf = pmod(fast + dfast, TWO_PI_F);
        if (crossed) nf = pmod(nf * keep, TWO_PI_F);
        prev = slow; slow = ns; fast = nf;
    }
}

// out3[base + 16*i + j] = in[base + 16*i + j] * s, one 16x16 tile per wave,
// via 4 chained V_WMMA_F32_16X16X4_F32: D = sum_m A_m x B_m,
//   A_m[i][k] = in-tile[i][4m+k]        (documented 16x4 f32 A layout)
//   B_m[k][j] = s * delta_{j, 4m+k}     (B layout: lane l -> N = l&15,
//                                        K = 2*(l>>4) + vgpr, mirror of A)
// Exactly one non-zero product lands on each D element -> bit-exact scale.
__global__ void osc_scale_wmma_kernel(const float* __restrict__ in,
                                      float* __restrict__ out, int N) {
    const float s = INV_TWO_PI_F;
    const int base = blockIdx.x * 256;
    const int lane = threadIdx.x;          // wave32: 0..31
    const int i    = lane & 15;            // A-matrix M row / B N column / D N col
    const int kh   = lane >> 4;            // lane-half selects K pair

    if (base + 256 <= N) {
        v8f d = {0.f, 0.f, 0.f, 0.f, 0.f, 0.f, 0.f, 0.f};
#pragma unroll
        for (int m = 0; m < 4; ++m) {
            // A_m: lane holds T[i][4m + 2*kh + {0,1}] in its two A VGPRs.
            const float* p = in + base + i * 16 + 4 * m + 2 * kh;
            v2f a; a.x = p[0]; a.y = p[1];
            // B_m: s at (k, j=4m+k); this lane's j == i, k == 2*kh + vgpr.
            v2f b;
            b.x = (i == (4 * m + 2 * kh + 0)) ? s : 0.0f;
            b.y = (i == (4 * m + 2 * kh + 1)) ? s : 0.0f;
            // 8 args: (neg_a, A, neg_b, B, c_mod, C, reuse_a, reuse_b)
            d = __builtin_amdgcn_wmma_f32_16x16x4_f32(
                    false, a, false, b, (short)0, d, false, false);
        }
        // D layout: VGPR r, lanes 0-15 -> (M=r, N=lane); lanes 16-31 -> (M=r+8).
#pragma unroll
        for (int r = 0; r < 8; ++r) {
            out[base + (r + 8 * kh) * 16 + i] = d[r];
        }
    } else {
        // Partial tail tile: uniform branch, plain VALU scale.
        for (int j = base + lane; j < N; j += 32) out[j] = in[j] * s;
    }
}

extern "C" void kernel_launch(void* const* d_in, const int* in_sizes, int n_in,
                              void* d_out, int out_size, void* d_ws, size_t ws_size,
                              hipStream_t stream) {
    const float* p_log_slow = (const float*)d_in[0];
    const float* p_log_fast = (const float*)d_in[1];
    const float* p_reset    = (const float*)d_in[2];
    // d_in[3] holds `steps` on device; host derives N from out_size instead
    // (3 concatenated outputs of length steps) to stay graph-capture safe.
    const int N = out_size / 3;
    if (N <= 0) return;
    float* out = (float*)d_out;

    osc_scan_kernel<<<1, 32, 0, stream>>>(p_log_slow, p_log_fast, p_reset, out, N);

    const int tiles = (N + 255) / 256;
    osc_scale_wmma_kernel<<<tiles, 32, 0, stream>>>(out, out + 2 * (size_t)N, N);
}